// GraphSage_5317169512695
// MI455X (gfx1250) — compile-verified
//
#include <hip/hip_runtime.h>

#define NN 20000
#define EE 320000

typedef float v2f __attribute__((ext_vector_type(2)));
typedef float v4f __attribute__((ext_vector_type(4)));
typedef float v8f __attribute__((ext_vector_type(8)));

// ---------------------------------------------------------------- utility
__global__ void zero_i32(int* p, int n) {
    int i = blockIdx.x * 256 + threadIdx.x;
    if (i < n) p[i] = 0;
}

__global__ void count_deg(const long long* __restrict__ ei, int* __restrict__ deg) {
    int e = blockIdx.x * 256 + threadIdx.x;
    if (e >= EE) return;
    atomicAdd(&deg[(int)ei[(size_t)EE + e]], 1);
}

__global__ void make_deg_inv(const int* __restrict__ deg, float* __restrict__ dinv) {
    int n = blockIdx.x * 256 + threadIdx.x;
    if (n >= NN) return;
    dinv[n] = 1.0f / fmaxf((float)deg[n], 1.0f);
}

// single-block exclusive scan of deg -> rowptr (N=20000, 20 tiles of 1024)
__global__ void scan_deg(const int* __restrict__ deg, int* __restrict__ rowptr) {
    __shared__ int s[1024];
    __shared__ int carry;
    const int tid = threadIdx.x;
    if (tid == 0) carry = 0;
    __syncthreads();
    for (int base = 0; base < NN; base += 1024) {
        int v = (base + tid < NN) ? deg[base + tid] : 0;
        s[tid] = v;
        __syncthreads();
        for (int off = 1; off < 1024; off <<= 1) {
            int t = (tid >= off) ? s[tid - off] : 0;
            __syncthreads();
            s[tid] += t;
            __syncthreads();
        }
        if (base + tid < NN) rowptr[base + tid] = carry + s[tid] - v;  // exclusive
        __syncthreads();
        if (tid == 1023) carry += s[1023];
        __syncthreads();
    }
    if (tid == 0) rowptr[NN] = carry;  // == EE
}

__global__ void fill_csr(const long long* __restrict__ ei, const int* __restrict__ rowptr,
                         int* __restrict__ cursor, int* __restrict__ csr) {
    int e = blockIdx.x * 256 + threadIdx.x;
    if (e >= EE) return;
    int d = (int)ei[(size_t)EE + e];
    int pos = rowptr[d] + atomicAdd(&cursor[d], 1);
    csr[pos] = (int)ei[e];
}

// ------------------------------------------------- weight repack (fragment-major)
// Quad layout: one b128 per lane delivers the B fragments of TWO consecutive
// K-blocks (kb = 2*kb2 and 2*kb2+1) for column tile ct:
//   P[q*4 + {0,1}] = W[kb2*8 + 2*(lane>>4) + {0,1}    ][ct*16 + (lane&15)]
//   P[q*4 + {2,3}] = W[kb2*8 + 2*(lane>>4) + {0,1} + 4][ct*16 + (lane&15)]
__global__ void pack_weights(const float* __restrict__ W, float* __restrict__ P,
                             int fi, int fo) {
    int q = blockIdx.x * 256 + threadIdx.x;
    int total = (fi * fo) >> 2;
    if (q >= total) return;
    int lane = q & 31;
    int tmp  = q >> 5;
    int foT  = fo >> 4;
    int ct   = tmp % foT;
    int kb2  = tmp / foT;
    int col  = (ct << 4) + (lane & 15);
    size_t kbase = (size_t)(kb2 << 3) + ((lane >> 4) << 1);
    P[4 * q + 0] = W[(kbase + 0) * fo + col];
    P[4 * q + 1] = W[(kbase + 1) * fo + col];
    P[4 * q + 2] = W[(kbase + 4) * fo + col];
    P[4 * q + 3] = W[(kbase + 5) * fo + col];
}

// ------------------------------------------------- mean aggregation (CSR gather)
// blockDim chosen per-layer so every lane owns a 16B (float4) slice: F=1536 ->
// 192 thr x 2 chunks; 768 -> 192 thr; 384 -> 96 thr; 256 -> 64 thr.
__global__ void sage_aggregate(const float* __restrict__ X, float* __restrict__ Agg,
                               const int* __restrict__ rowptr, const int* __restrict__ csr,
                               const float* __restrict__ dinv, int F) {
    const int n      = blockIdx.x;
    const int f0     = threadIdx.x << 2;
    const int stride = blockDim.x << 2;
    const int nc     = F / stride;  // 1 or 2
    v4f acc0 = (v4f){0.f, 0.f, 0.f, 0.f};
    v4f acc1 = acc0;
    const int beg = rowptr[n], end = rowptr[n + 1];
    for (int e = beg; e < end; ++e) {
        const float* xs = X + (size_t)csr[e] * F;
        if (e + 1 < end)  // gfx1250 global_prefetch_b8 of next neighbor row
            __builtin_prefetch(X + (size_t)csr[e + 1] * F + f0, 0, 0);
        acc0 += *(const v4f*)(xs + f0);
        if (nc > 1) acc1 += *(const v4f*)(xs + f0 + stride);
    }
    const float di = dinv[n];
    float* out = Agg + (size_t)n * F;
    *(v4f*)(out + f0) = acc0 * di;
    if (nc > 1) *(v4f*)(out + f0 + stride) = acc1 * di;
}

// ------------------------------------------------- fused dual GEMM via fp32 WMMA
// H = relu?(Agg @ Wl + X @ Wr + b). Wave: 32 rows x 64 cols, K-step 8 (2 WMMA K-blocks).
__global__ __launch_bounds__(256) void sage_gemm(
    const float* __restrict__ Agg, const float* __restrict__ X,
    const float* __restrict__ pWl, const float* __restrict__ pWr,
    const float* __restrict__ bias, float* __restrict__ H,
    int Fi, int Fo, int doRelu) {
    const int lane = threadIdx.x & 31;
    const int wave = threadIdx.x >> 5;
    const int row0 = blockIdx.x << 5;                    // 32 rows per block
    const int colw = (blockIdx.y << 9) + (wave << 6);    // 64 cols per wave
    if (colw >= Fo) return;

    const int m  = lane & 15;
    const int kh = lane >> 4;  // half-lane selector

    // A fragments: lane loads float2 at [row0+m][4*kb + 2*kh .. +1]
    const float* a0 = Agg + (size_t)(row0 + m) * Fi + (kh << 1);
    const float* a1 = a0 + (size_t)16 * Fi;
    const float* x0 = X   + (size_t)(row0 + m) * Fi + (kh << 1);
    const float* x1 = x0 + (size_t)16 * Fi;

    const int foT = Fo >> 4;
    const v4f* bl = (const v4f*)pWl + (size_t)(colw >> 4) * 32 + lane;
    const v4f* br = (const v4f*)pWr + (size_t)(colw >> 4) * 32 + lane;
    const size_t bStep = (size_t)foT * 32;  // v4f elements per kb2

    v8f acc[2][4];
#pragma unroll
    for (int i = 0; i < 2; ++i)
#pragma unroll
        for (int t = 0; t < 4; ++t)
            acc[i][t] = (v8f){0.f, 0.f, 0.f, 0.f, 0.f, 0.f, 0.f, 0.f};

    const int KB2 = Fi >> 3;
    for (int kb2 = 0; kb2 < KB2; ++kb2) {
        const int k0 = kb2 << 3;
        v2f aA0a = *(const v2f*)(a0 + k0);
        v2f aA0b = *(const v2f*)(a0 + k0 + 4);
        v2f aA1a = *(const v2f*)(a1 + k0);
        v2f aA1b = *(const v2f*)(a1 + k0 + 4);
        v2f aX0a = *(const v2f*)(x0 + k0);
        v2f aX0b = *(const v2f*)(x0 + k0 + 4);
        v2f aX1a = *(const v2f*)(x1 + k0);
        v2f aX1b = *(const v2f*)(x1 + k0 + 4);
#pragma unroll
        for (int t = 0; t < 4; ++t) {
            v4f wl4 = bl[t * 32];
            v4f wr4 = br[t * 32];
            v2f wla = (v2f){wl4.x, wl4.y};
            v2f wlb = (v2f){wl4.z, wl4.w};
            v2f wra = (v2f){wr4.x, wr4.y};
            v2f wrb = (v2f){wr4.z, wr4.w};
            acc[0][t] = __builtin_amdgcn_wmma_f32_16x16x4_f32(
                false, aA0a, false, wla, (short)0, acc[0][t], false, false);
            acc[1][t] = __builtin_amdgcn_wmma_f32_16x16x4_f32(
                false, aA1a, false, wla, (short)0, acc[1][t], false, false);
            acc[0][t] = __builtin_amdgcn_wmma_f32_16x16x4_f32(
                false, aX0a, false, wra, (short)0, acc[0][t], false, false);
            acc[1][t] = __builtin_amdgcn_wmma_f32_16x16x4_f32(
                false, aX1a, false, wra, (short)0, acc[1][t], false, false);
            acc[0][t] = __builtin_amdgcn_wmma_f32_16x16x4_f32(
                false, aA0b, false, wlb, (short)0, acc[0][t], false, false);
            acc[1][t] = __builtin_amdgcn_wmma_f32_16x16x4_f32(
                false, aA1b, false, wlb, (short)0, acc[1][t], false, false);
            acc[0][t] = __builtin_amdgcn_wmma_f32_16x16x4_f32(
                false, aX0b, false, wrb, (short)0, acc[0][t], false, false);
            acc[1][t] = __builtin_amdgcn_wmma_f32_16x16x4_f32(
                false, aX1b, false, wrb, (short)0, acc[1][t], false, false);
        }
        bl += bStep;
        br += bStep;
    }

    // C/D layout: vgpr r -> row = r + 8*kh; col = lane&15
#pragma unroll
    for (int t = 0; t < 4; ++t) {
        const int col = colw + (t << 4) + m;
        const float bv = bias[col];
#pragma unroll
        for (int r = 0; r < 8; ++r) {
            const int row = row0 + (kh << 3) + r;
            float v0 = acc[0][t][r] + bv;
            float v1 = acc[1][t][r] + bv;
            if (doRelu) { v0 = fmaxf(v0, 0.f); v1 = fmaxf(v1, 0.f); }
            H[(size_t)row * Fo + col] = v0;
            H[(size_t)(row + 16) * Fo + col] = v1;
        }
    }
}

// ------------------------------------------------- final tiny layer (Fo = 5)
__global__ void out_layer5(const float* __restrict__ agg, const float* __restrict__ h,
                           const float* __restrict__ Wl, const float* __restrict__ Wr,
                           const float* __restrict__ b, float* __restrict__ out) {
    int idx = blockIdx.x * 256 + threadIdx.x;
    if (idx >= NN * 5) return;
    int n = idx / 5, j = idx % 5;
    const float* ag = agg + (size_t)n * 256;
    const float* hh = h + (size_t)n * 256;
    float s = b[j];
    for (int k = 0; k < 256; ++k)
        s += ag[k] * Wl[k * 5 + j] + hh[k] * Wr[k * 5 + j];
    out[idx] = s;
}

// ----------------------------------------------------------------- driver
extern "C" void kernel_launch(void* const* d_in, const int* in_sizes, int n_in,
                              void* d_out, int out_size, void* d_ws, size_t ws_size,
                              hipStream_t stream) {
    const float* x = (const float*)d_in[0];
    const long long* ei = (const long long*)d_in[1];
    const float* Wl[5], *Wr[5], *bb[5];
    for (int l = 0; l < 5; ++l) {
        Wl[l] = (const float*)d_in[2 + 3 * l];
        Wr[l] = (const float*)d_in[3 + 3 * l];
        bb[l] = (const float*)d_in[4 + 3 * l];
    }
    float* outH = (float*)d_out;               // [N,256]
    float* outY = outH + (size_t)NN * 256;     // [N,5]

    // workspace carve-out (256B aligned)
    char* p = (char*)d_ws;
    auto carve = [&](size_t bytes) -> void* {
        void* r = (void*)p;
        p += (bytes + 255) & ~(size_t)255;
        return r;
    };
    int*   deg    = (int*)  carve(NN * 4);
    int*   cursor = (int*)  carve(NN * 4);
    int*   rowptr = (int*)  carve((NN + 1) * 4);
    int*   csr    = (int*)  carve((size_t)EE * 4);
    float* dinv   = (float*)carve(NN * 4);
    float* aggbuf = (float*)carve((size_t)NN * 1536 * 4);
    float* bufA   = (float*)carve((size_t)NN * 1536 * 4);  // h1
    float* bufB   = (float*)carve((size_t)NN * 768 * 4);   // h2
    float* bufC   = (float*)carve((size_t)NN * 384 * 4);   // h3
    const int FI[4] = {768, 1536, 768, 384};
    const int FO[4] = {1536, 768, 384, 256};
    float* pWl[4], *pWr[4];
    for (int l = 0; l < 4; ++l) {
        pWl[l] = (float*)carve((size_t)FI[l] * FO[l] * 4);
        pWr[l] = (float*)carve((size_t)FI[l] * FO[l] * 4);
    }

    // ---- graph preprocessing (reused by all 5 layers)
    zero_i32<<<(NN + 255) / 256, 256, 0, stream>>>(deg, NN);
    zero_i32<<<(NN + 255) / 256, 256, 0, stream>>>(cursor, NN);
    count_deg<<<(EE + 255) / 256, 256, 0, stream>>>(ei, deg);
    make_deg_inv<<<(NN + 255) / 256, 256, 0, stream>>>(deg, dinv);
    scan_deg<<<1, 1024, 0, stream>>>(deg, rowptr);
    fill_csr<<<(EE + 255) / 256, 256, 0, stream>>>(ei, rowptr, cursor, csr);

    // ---- repack weights into WMMA fragment-major (paired-K b128) layout
    for (int l = 0; l < 4; ++l) {
        int tot = (FI[l] * FO[l]) >> 2;
        pack_weights<<<(tot + 255) / 256, 256, 0, stream>>>(Wl[l], pWl[l], FI[l], FO[l]);
        pack_weights<<<(tot + 255) / 256, 256, 0, stream>>>(Wr[l], pWr[l], FI[l], FO[l]);
    }

    const float* xin[4] = {x, bufA, bufB, bufC};
    float* hout[4] = {bufA, bufB, bufC, outH};

    for (int l = 0; l < 4; ++l) {
        int thr = (FI[l] / 4 > 256) ? 192 : FI[l] / 4;  // 1536->192x2, 768->192, 384->96, 256->64
        sage_aggregate<<<NN, thr, 0, stream>>>(xin[l], aggbuf, rowptr, csr, dinv, FI[l]);
        dim3 grid(NN / 32, (FO[l] + 511) >> 9);
        sage_gemm<<<grid, 256, 0, stream>>>(aggbuf, xin[l], pWl[l], pWr[l], bb[l],
                                            hout[l], FI[l], FO[l], 1);
    }

    // layer 5: aggregate h4 (in d_out) then tiny dense layer, no relu
    sage_aggregate<<<NN, 64, 0, stream>>>(outH, aggbuf, rowptr, csr, dinv, 256);
    out_layer5<<<(NN * 5 + 255) / 256, 256, 0, stream>>>(aggbuf, outH, Wl[4], Wr[4],
                                                         bb[4], outY);
}